// S6Core_27934467293643
// MI455X (gfx1250) — compile-verified
//
#include <hip/hip_runtime.h>
#include <hip/hip_bf16.h>
#include <stdint.h>

// ---------------------------------------------------------------------------
// S6 selective scan (Mamba core) for gfx1250 (MI455X, wave32, WMMA + TDM)
//   B=4, L=2048, D=1024, N=16, dt_rank=64
// Pipeline:
//   K1: x_dbl[8192][96]  = x @ W_xproj^T                  (v_wmma_f32_16x16x4_f32)
//   K2: dt[8192][1024]   = softplus(x_dbl[:,:64] @ W_dt^T + b_dt)   (WMMA)
//   K3: sequential scan; per-chunk operand tiles staged into LDS by the
//       Tensor Data Mover (tensor_load_to_lds, double buffered, TENSORcnt),
//       inner loop runs off LDS; y-reduce over N=16 via lane shuffles.
// ---------------------------------------------------------------------------

typedef __attribute__((ext_vector_type(2))) float v2f;
typedef __attribute__((ext_vector_type(8))) float v8f;
typedef __attribute__((ext_vector_type(4))) unsigned int u32x4;
typedef __attribute__((ext_vector_type(8))) int i32x8;
typedef __attribute__((ext_vector_type(4))) int i32x4;

#define DMODEL 1024
#define DSTATE 16
#define DTRANK 64
#define BATCH  4
#define SEQLEN 2048
#define MROWS  (BATCH * SEQLEN)      // 8192
#define KPROJ  (DTRANK + 2 * DSTATE) // 96
#define CHUNK  64                    // timesteps per TDM-staged LDS chunk
#define NCHUNK (SEQLEN / CHUNK)      // 32

// ---------------------------------------------------------------------------
// K1: x_dbl = x @ W_xproj^T.  M=8192, N=96, K=1024. 16x16 tile per wave.
// ---------------------------------------------------------------------------
__global__ __launch_bounds__(128) void s6_xproj_gemm(
    const float* __restrict__ X,   // [8192][1024]
    const float* __restrict__ W,   // [96][1024]
    float* __restrict__ XD)        // [8192][96]
{
    const int wave  = threadIdx.x >> 5;
    const int lane  = threadIdx.x & 31;
    const int mtile = (blockIdx.x * 4 + wave) * 16;
    const int ntile = blockIdx.y * 16;

    const int lm = lane & 15;          // M (A) / N (B) row within tile
    const int lk = (lane >> 4) << 1;   // K sub-offset: 0 or 2

    const float* arow = X + (size_t)(mtile + lm) * DMODEL + lk;
    const float* brow = W + (size_t)(ntile + lm) * DMODEL + lk;

    v8f c = {};
    #pragma unroll 8
    for (int k = 0; k < DMODEL; k += 4) {
        v2f a = *(const v2f*)(arow + k);
        v2f b = *(const v2f*)(brow + k);
        c = __builtin_amdgcn_wmma_f32_16x16x4_f32(
                false, a, false, b, (short)0, c, false, false);
    }

    const int row0 = mtile + ((lane >> 4) << 3);
    float* out = XD + (size_t)row0 * KPROJ + ntile + lm;
    #pragma unroll
    for (int i = 0; i < 8; ++i) out[(size_t)i * KPROJ] = c[i];
}

// ---------------------------------------------------------------------------
// K2: dt = softplus(x_dbl[:, :64] @ W_dt^T + b_dt).  M=8192, N=1024, K=64.
// ---------------------------------------------------------------------------
__global__ __launch_bounds__(128) void s6_dt_gemm(
    const float* __restrict__ XD,   // [8192][96] (cols 0..63 used)
    const float* __restrict__ Wdt,  // [1024][64]
    const float* __restrict__ bdt,  // [1024]
    float* __restrict__ DT)         // [8192][1024]
{
    const int wave  = threadIdx.x >> 5;
    const int lane  = threadIdx.x & 31;
    const int mtile = (blockIdx.x * 4 + wave) * 16;
    const int ntile = blockIdx.y * 16;

    const int lm = lane & 15;
    const int lk = (lane >> 4) << 1;

    const float* arow = XD  + (size_t)(mtile + lm) * KPROJ  + lk;
    const float* brow = Wdt + (size_t)(ntile + lm) * DTRANK + lk;

    v8f c = {};
    #pragma unroll
    for (int k = 0; k < DTRANK; k += 4) {
        v2f a = *(const v2f*)(arow + k);
        v2f b = *(const v2f*)(brow + k);
        c = __builtin_amdgcn_wmma_f32_16x16x4_f32(
                false, a, false, b, (short)0, c, false, false);
    }

    const int n    = ntile + lm;
    const float bb = bdt[n];
    const int row0 = mtile + ((lane >> 4) << 3);
    float* out = DT + (size_t)row0 * DMODEL + n;
    #pragma unroll
    for (int i = 0; i < 8; ++i) {
        float z  = c[i] + bb;
        float sp = (z > 20.0f) ? z : __logf(1.0f + __expf(z));
        out[(size_t)i * DMODEL] = sp;
    }
}

// ---------------------------------------------------------------------------
// Tensor Data Mover: 2D tile (rows x width f32, row stride in elements)
// from global memory into LDS at lds_off.  D# per CDNA5 ISA §8.3/8.4.
// ---------------------------------------------------------------------------
__device__ __forceinline__ void tdm_load_2d(uint32_t lds_off, const void* gsrc,
                                            uint32_t width_elts, uint32_t rows,
                                            uint64_t row_stride_elts)
{
    const uint64_t ga = (uint64_t)(uintptr_t)gsrc;

    u32x4 g0;
    g0[0] = 1u;                                            // count=1, user D#
    g0[1] = lds_off;                                       // lds_addr (bytes)
    g0[2] = (uint32_t)(ga & 0xFFFFFFFFu);                  // global_addr[31:0]
    g0[3] = (uint32_t)((ga >> 32) & 0x1FFFFFFu)            // global_addr[56:32]
          | (2u << 30);                                    // type = 2 (image)

    // group1 (256b) assembled as four 64-bit quads
    const uint64_t q0 = (2ull << 16)                              // data_size=4B
                      | ((uint64_t)(width_elts & 0xFFFFu) << 48); // tensor_dim0 lo @48
    const uint64_t q1 = ((uint64_t)width_elts >> 16)              // tensor_dim0 hi
                      | ((uint64_t)rows << 16)                    // tensor_dim1 @80
                      | ((uint64_t)(width_elts & 0xFFFFu) << 48); // tile_dim0 @112
    const uint64_t q2 = (uint64_t)(rows & 0xFFFFu)                // tile_dim1 @128
                      | ((row_stride_elts & 0xFFFFFFFFull) << 32);// dim0_stride lo @160
    const uint64_t q3 = (row_stride_elts >> 32) & 0xFFFFull;      // dim0_stride hi

    i32x8 g1;
    g1[0] = (int)(uint32_t)q0;  g1[1] = (int)(uint32_t)(q0 >> 32);
    g1[2] = (int)(uint32_t)q1;  g1[3] = (int)(uint32_t)(q1 >> 32);
    g1[4] = (int)(uint32_t)q2;  g1[5] = (int)(uint32_t)(q2 >> 32);
    g1[6] = (int)(uint32_t)q3;  g1[7] = (int)(uint32_t)(q3 >> 32);

    const i32x4 gz = {0, 0, 0, 0};
#if defined(__clang_major__) && (__clang_major__ >= 23)
    const i32x8 gz8 = {0, 0, 0, 0, 0, 0, 0, 0};
    __builtin_amdgcn_tensor_load_to_lds(g0, g1, gz, gz, gz8, 0);
#else
    __builtin_amdgcn_tensor_load_to_lds(g0, g1, gz, gz, 0);
#endif
}

// ---------------------------------------------------------------------------
// K3: sequential scan with TDM-staged, double-buffered LDS operand tiles.
// Block: 32 channels x 16 states = 512 threads (16 waves); grid (B, D/32).
// Wave 0 drives the TDM; TENSORcnt + workgroup barriers synchronize.
// ---------------------------------------------------------------------------
__global__ __launch_bounds__(512) void s6_scan(
    const float* __restrict__ X,     // [B][L][D]
    const float* __restrict__ XD,    // [8192][96]: B_sel 64..79, C_sel 80..95
    const float* __restrict__ DT,    // [B][L][D]
    const float* __restrict__ Alog,  // [D][N]
    const float* __restrict__ Dp,    // [D]
    float* __restrict__ Y)           // [B][L][D]
{
    __shared__ float bc_buf[2][CHUNK][32];  // B (cols 0..15), C (cols 16..31)
    __shared__ float dt_buf[2][CHUNK][32];
    __shared__ float x_buf [2][CHUNK][32];

    const int b    = blockIdx.x;
    const int d0   = blockIdx.y * 32;
    const int n    = threadIdx.x & 15;
    const int dloc = threadIdx.x >> 4;          // 0..31
    const int d    = d0 + dloc;

    const float A    = -__expf(Alog[d * DSTATE + n]);
    const float Dpar = Dp[d];

    const bool issuer = (threadIdx.x < 32);     // wave 0 only drives TDM

    auto issue_chunk = [&](int c, int buf) {
        const size_t row0 = (size_t)b * SEQLEN + (size_t)c * CHUNK;
        tdm_load_2d((uint32_t)(uintptr_t)&bc_buf[buf][0][0],
                    XD + row0 * KPROJ + DTRANK, 32, CHUNK, KPROJ);
        tdm_load_2d((uint32_t)(uintptr_t)&dt_buf[buf][0][0],
                    DT + row0 * DMODEL + d0, 32, CHUNK, DMODEL);
        tdm_load_2d((uint32_t)(uintptr_t)&x_buf[buf][0][0],
                    X + row0 * DMODEL + d0, 32, CHUNK, DMODEL);
    };

    if (issuer) issue_chunk(0, 0);

    float*       yb = Y + ((size_t)b * SEQLEN) * DMODEL + d;
    float h = 0.0f;

    for (int c = 0; c < NCHUNK; ++c) {
        if (issuer) {
            if (c + 1 < NCHUNK) {
                issue_chunk(c + 1, (c + 1) & 1);          // prefetch next chunk
                __builtin_amdgcn_s_wait_tensorcnt(3);     // chunk c's 3 ops done
            } else {
                __builtin_amdgcn_s_wait_tensorcnt(0);
            }
        }
        __syncthreads();                                  // publish LDS tiles

        const int buf = c & 1;
        const int tbase = c * CHUNK;
        #pragma unroll 4
        for (int tt = 0; tt < CHUNK; ++tt) {
            const float dt = dt_buf[buf][tt][dloc];       // ds broadcast per group
            const float xv = x_buf [buf][tt][dloc];
            const float Bt = bc_buf[buf][tt][n];
            const float Ct = bc_buf[buf][tt][16 + n];

            const float abar = __expf(dt * A);            // off the h chain
            h = fmaf(abar, h, dt * xv * Bt);              // serial dependency

            float p = h * Ct;                             // reduce over 16 lanes
            p += __shfl_xor(p, 8, 16);
            p += __shfl_xor(p, 4, 16);
            p += __shfl_xor(p, 2, 16);
            p += __shfl_xor(p, 1, 16);
            if (n == 0)
                yb[(size_t)(tbase + tt) * DMODEL] = fmaf(xv, Dpar, p);
        }
        __syncthreads();   // all reads of buf done before it is re-written
    }
}

// ---------------------------------------------------------------------------
extern "C" void kernel_launch(void* const* d_in, const int* in_sizes, int n_in,
                              void* d_out, int out_size, void* d_ws, size_t ws_size,
                              hipStream_t stream)
{
    const float* x    = (const float*)d_in[0];  // [4][2048][1024]
    const float* Wx   = (const float*)d_in[1];  // [96][1024]
    const float* Wdt  = (const float*)d_in[2];  // [1024][64]
    const float* bdt  = (const float*)d_in[3];  // [1024]
    const float* Alog = (const float*)d_in[4];  // [1024][16]
    const float* Dp   = (const float*)d_in[5];  // [1024]
    float* Y = (float*)d_out;

    float* XD = (float*)d_ws;                                        // 3 MB
    float* DT = (float*)((char*)d_ws + (size_t)MROWS * KPROJ * 4);   // 33.5 MB

    s6_xproj_gemm<<<dim3(128, 6), 128, 0, stream>>>(x, Wx, XD);
    s6_dt_gemm<<<dim3(128, 64), 128, 0, stream>>>(XD, Wdt, bdt, DT);
    s6_scan<<<dim3(BATCH, DMODEL / 32), 512, 0, stream>>>(x, XD, DT, Alog, Dp, Y);
}